// SpatioTemporalGC_41369124995800
// MI455X (gfx1250) — compile-verified
//
#include <hip/hip_runtime.h>
#include <stdint.h>

#ifndef __has_builtin
#define __has_builtin(x) 0
#endif

namespace {
constexpr uint32_t T      = 50;           // SEQ_LEN
constexpr uint32_t NCH    = 66;           // N_CHANNELS
constexpr uint32_t NROWS  = NCH * T;      // 3300 band rows (period of i mod 3300)
constexpr uint32_t TOTAL  = 108134400u;   // 32768 * 66 * 50
constexpr uint32_t TPB    = 256;
constexpr uint32_t TILE   = 1024;         // elements per block (TPB * 4)
constexpr uint32_t NBLK   = TOTAL / TILE; // 105600, exact
}

// ---- CDNA5 async global->LDS copy (ASYNCcnt path) -------------------------
__device__ __forceinline__ void async_load_b128_to_lds(const float* gsrc,
                                                       uint32_t lds_byte_off) {
  // vdst = VGPR holding LDS byte address; vaddr = 64-bit per-lane address.
  asm volatile("global_load_async_to_lds_b128 %0, %1, off"
               :: "v"(lds_byte_off), "v"(gsrc)
               : "memory");
}

__device__ __forceinline__ void wait_async0() {
#if __has_builtin(__builtin_amdgcn_s_wait_asynccnt)
  __builtin_amdgcn_s_wait_asynccnt(0);
#else
  asm volatile("s_wait_asynccnt 0x0" ::: "memory");
#endif
}

// ---- Main streaming banded-matvec kernel (hot path, defined FIRST so the
// ---- disasm snippet shows it). w = compact band[m][3], m = i mod 3300. ----
__global__ void __launch_bounds__(TPB)
stgc_banded_band_kernel(const float* __restrict__ x,
                        const float* __restrict__ w,
                        float* __restrict__ out) {
  // xs[3] = left halo, xs[4..4+TILE-1] = tile, xs[4+TILE] = right halo.
  // Tile start at xs[4] keeps async b128 destinations 16B-aligned.
  __shared__ __align__(16) float xs[TILE + 8];

  const uint32_t tid   = threadIdx.x;
  const uint32_t gbase = (uint32_t)blockIdx.x * TILE;

  // Stage the 4KB x-tile via the CDNA5 async path (ASYNCcnt tracked).
  async_load_b128_to_lds(x + (size_t)gbase + tid * 4u,
                         (uint32_t)(uintptr_t)&xs[4u + tid * 4u]);

  // Halos (band zero-padding kills their weights at true row edges; only the
  // global array ends need guarding).
  if (tid == 0) xs[3] = (gbase == 0u) ? 0.0f : x[(size_t)gbase - 1u];
  if (tid == 1) xs[4u + TILE] =
      (gbase + TILE >= TOTAL) ? 0.0f : x[(size_t)gbase + TILE];

  // Block-uniform band phase: computed once on the SALU, no per-lane div.
  // i = b*3300 + (n*50 + f)  =>  band row m = i mod 3300.
  const uint32_t s = gbase % NROWS;       // scalar (uniform in blockIdx)

  wait_async0();      // this wave's async transfers landed in LDS
  __syncthreads();    // all waves' transfers visible block-wide

  // 6 contiguous LDS reads cover the 3-tap windows of 4 outputs.
  float xv[6];
#pragma unroll
  for (int k = 0; k < 6; ++k) xv[k] = xs[3u + tid * 4u + (uint32_t)k];

  uint32_t m = s + tid * 4u;              // < 3300 + 1020
  if (m >= NROWS) m -= NROWS;             // single conditional subtract

  const uint32_t i0 = gbase + tid * 4u;
  float4 r;
  float* rp = &r.x;
#pragma unroll
  for (int j = 0; j < 4; ++j) {
    uint32_t mj = m + (uint32_t)j;        // < 3303
    if (mj >= NROWS) mj -= NROWS;         // wrap at batch boundary
    const float* bp = w + (size_t)mj * 3u;     // contiguous, L2-hot
    float b0 = bp[0], b1 = bp[1], b2 = bp[2];
    rp[j] = fmaf(b0, xv[j], fmaf(b1, xv[j + 1], b2 * xv[j + 2]));
  }
  *reinterpret_cast<float4*>(out + (size_t)i0) = r;  // coalesced b128 store
}

// ---- Phase 0: compact the tridiagonal band of adj into band[n*50+f][3] ----
__global__ void band_repack_kernel(const float* __restrict__ adj,
                                   float* __restrict__ band) {
  uint32_t j = blockIdx.x * blockDim.x + threadIdx.x;  // 0..3299
  if (j >= NROWS) return;
  uint32_t n = j / T;
  uint32_t f = j - n * T;
  const float* a = adj + (size_t)n * T * T + (size_t)f * T;
  float b0 = (f > 0)     ? a[f - 1] : 0.0f;  // masked left diag
  float b1 = a[f];                            // main diag
  float b2 = (f < T - 1) ? a[f + 1] : 0.0f;  // masked right diag (zero-pad)
  band[3u * j + 0u] = b0;
  band[3u * j + 1u] = b1;
  band[3u * j + 2u] = b2;
}

// ---- Fallback (only if ws too small): raw adj reads, div-based cold path --
__global__ void __launch_bounds__(TPB)
stgc_banded_raw_kernel(const float* __restrict__ x,
                       const float* __restrict__ adj,
                       float* __restrict__ out) {
  __shared__ __align__(16) float xs[TILE + 8];

  const uint32_t tid   = threadIdx.x;
  const uint32_t gbase = (uint32_t)blockIdx.x * TILE;

  async_load_b128_to_lds(x + (size_t)gbase + tid * 4u,
                         (uint32_t)(uintptr_t)&xs[4u + tid * 4u]);
  if (tid == 0) xs[3] = (gbase == 0u) ? 0.0f : x[(size_t)gbase - 1u];
  if (tid == 1) xs[4u + TILE] =
      (gbase + TILE >= TOTAL) ? 0.0f : x[(size_t)gbase + TILE];

  const uint32_t s = gbase % NROWS;

  wait_async0();
  __syncthreads();

  float xv[6];
#pragma unroll
  for (int k = 0; k < 6; ++k) xv[k] = xs[3u + tid * 4u + (uint32_t)k];

  uint32_t m = s + tid * 4u;
  if (m >= NROWS) m -= NROWS;

  const uint32_t i0 = gbase + tid * 4u;
  float4 r;
  float* rp = &r.x;
#pragma unroll
  for (int j = 0; j < 4; ++j) {
    uint32_t mj = m + (uint32_t)j;
    if (mj >= NROWS) mj -= NROWS;
    uint32_t n = mj / T;
    uint32_t f = mj - n * T;
    const float* a = adj + (size_t)n * (T * T) + (size_t)f * T;
    float b0 = (f > 0)     ? a[f - 1] : 0.0f;
    float b1 = a[f];
    float b2 = (f < T - 1) ? a[f + 1] : 0.0f;
    rp[j] = fmaf(b0, xv[j], fmaf(b1, xv[j + 1], b2 * xv[j + 2]));
  }
  *reinterpret_cast<float4*>(out + (size_t)i0) = r;
}

extern "C" void kernel_launch(void* const* d_in, const int* in_sizes, int n_in,
                              void* d_out, int out_size, void* d_ws, size_t ws_size,
                              hipStream_t stream) {
  const float* x   = (const float*)d_in[0];   // [32768, 66, 50] f32
  const float* adj = (const float*)d_in[1];   // [66, 50, 50] f32
  float* out       = (float*)d_out;           // [32768, 66, 50] f32

  const size_t band_bytes = (size_t)NROWS * 3u * sizeof(float);  // 39.6 KB
  if (ws_size >= band_bytes) {
    float* band = (float*)d_ws;
    band_repack_kernel<<<(NROWS + 255) / 256, 256, 0, stream>>>(adj, band);
    stgc_banded_band_kernel<<<NBLK, TPB, 0, stream>>>(x, band, out);
  } else {
    stgc_banded_raw_kernel<<<NBLK, TPB, 0, stream>>>(x, adj, out);
  }
}